// PowerSpectrumLoss_42752104464642
// MI455X (gfx1250) — compile-verified
//
#include <hip/hip_runtime.h>
#include <math.h>
#include <stdint.h>

// ---------------------------------------------------------------------------
// PowerSpectrumLoss on MI455X (gfx1250):
//   FFT2 as Z = F * X * F (F = symmetric 256-pt DFT matrix), all matmuls via
//   V_WMMA_F32_16X16X4_F32 (memory-bound problem: 268 MB in -> ~11.5 us HBM
//   floor at 23.3 TB/s; only ~206 GFLOP -> keep full fp32 precision).
//   One workgroup per image; 320 KB WGP LDS holds a pair-planar staged F
//   column-group (128 KB, staged by the Tensor Data Mover) and the complex
//   intermediate V = X*Fg (128 KB).  Pair-planar layouts make every WMMA
//   operand a single b64 load (no VALU repacking).  Radial binning via
//   ds_add_f32 LDS atomics; per-image bins stored without global atomics.
// ---------------------------------------------------------------------------

typedef __attribute__((ext_vector_type(2))) float v2f;
typedef __attribute__((ext_vector_type(8))) float v8f;
typedef __attribute__((ext_vector_type(4))) unsigned int u32x4;
typedef __attribute__((ext_vector_type(8))) int i32x8;
typedef __attribute__((ext_vector_type(4))) int i32x4;

#define NPIX   256
#define NBINS  64
#define NSEG   66          // segments 0..65, valid bins are 1..64
#define GROUPW 64          // output columns per LDS group
#define NGROUP (NPIX / GROUPW)
#define GRP_FLOATS 32768   // floats per packed F group (re plane + im plane)

#define USE_TDM __has_builtin(__builtin_amdgcn_tensor_load_to_lds)

// RAD_PER_PIX = (5*60/256) * pi/180 ; ps2d scale = RAD^2 / (N*N * TSZ^2)
static constexpr double kPi   = 3.14159265358979323846;
static constexpr double kRad  = (5.0 * 60.0 / 256.0) * kPi / 180.0;
static constexpr double kPsSc = (kRad * kRad) / (65536.0 * 1.0e12);

// D = A(16x4) * B(4x16) + C --- fp32 WMMA, wave32.
// A per-lane: lanes 0-15 row M=lane, K={kk,kk+1}; lanes 16-31 K={kk+2,kk+3}.
static __device__ inline v8f wmma4(v2f a, v2f b, v8f c) {
    return __builtin_amdgcn_wmma_f32_16x16x4_f32(
        false, a, false, b, (short)0, c, false, false);
}

// ---------------------------------------------------------------------------
// Kernel 0: build DFT matrix F[k][m] = exp(-2*pi*i*k*m/256) in two layouts:
//  - planar Fre/Fim/Fimn[row][k]  (stage-2 A operands: b64 = {k,k+1} pair)
//  - Fpack[g]: per-group pair-planar LDS image {re plane | im plane},
//    plane[k/2][c][2]  (stage-1 B operands: b64 = {F[k][c],F[k+1][c]})
// ---------------------------------------------------------------------------
__global__ void build_dft_kernel(float* __restrict__ Fre, float* __restrict__ Fim,
                                 float* __restrict__ Fimn, float* __restrict__ Fpack) {
    const int k = blockIdx.x;
    const int m = threadIdx.x;
    const float t = (float)((k * m) & 255) * (float)(2.0 * kPi / 256.0);
    float s, c;
    __sincosf(t, &s, &c);
    const float re = c, im = -s;
    Fre [k * NPIX + m] = re;
    Fim [k * NPIX + m] = im;
    Fimn[k * NPIX + m] = -im;
    const int g = m >> 6, cc = m & 63, p = k >> 1, o = k & 1;
    float* base = Fpack + g * GRP_FLOATS;
    base[               (p * GROUPW + cc) * 2 + o] = re;
    base[GRP_FLOATS / 2 + (p * GROUPW + cc) * 2 + o] = im;
}

// ---------------------------------------------------------------------------
// Kernel 1: one block per image.  Z = F * (X * F), |Z|^2 -> LDS bin sums.
// ---------------------------------------------------------------------------
__global__ __launch_bounds__(256, 1)
void fft_ps_kernel(const float* __restrict__ pred,
                   const int*   __restrict__ seg,
                   const float* __restrict__ Fre,
                   const float* __restrict__ Fim,
                   const float* __restrict__ Fimn,
                   const float* __restrict__ Fpack,
                   float* __restrict__ binsum) {
    extern __shared__ char smem[];
    float* Fsre = (float*)smem;                   // [128][64][2]  64 KB
    float* Fsim = Fsre + GRP_FLOATS / 2;          // [128][64][2]  64 KB
    float* Vsre = (float*)(smem + 131072);        // [128][64][2]  64 KB
    float* Vsim = Vsre + GRP_FLOATS / 2;          // [128][64][2]  64 KB
    float* bins = (float*)(smem + 262144);        // [66] radial bin sums

    const int tid = threadIdx.x;
    const int wv  = tid >> 5;           // wave id 0..7
    const int ln  = tid & 31;
    const int lm  = ln & 15;            // M / N position within tile
    const int hi  = (ln >> 4) & 1;      // lane half: K offset +2, M offset +8
    const int b   = blockIdx.x;
    const float* X = pred + (size_t)b * (NPIX * NPIX);

    if (tid < NSEG) bins[tid] = 0.0f;
    // warm the image through L2 (global_prefetch_b8)
    for (int i = tid * 16; i < NPIX * NPIX; i += 256 * 16)
        __builtin_prefetch(X + i, 0, 3);
    __syncthreads();

    for (int g = 0; g < NGROUP; ++g) {
        // ---- stage packed F group (128 KB) into LDS ----
#if USE_TDM
        if (tid < 32) {   // one wave drives the Tensor Data Mover
            const unsigned long long gaddr =
                (unsigned long long)(uintptr_t)(Fpack + g * GRP_FLOATS);
            const unsigned lds_off = (unsigned)(uintptr_t)Fsre;
            u32x4 d0;
            d0[0] = 1u;                                   // count=1, user mode
            d0[1] = lds_off;                              // lds_addr
            d0[2] = (unsigned)(gaddr & 0xffffffffull);    // global_addr[31:0]
            d0[3] = (unsigned)((gaddr >> 32) & 0x1ffffffull) | (2u << 30); // [56:32] | type=2
            i32x8 d1;
            d1[0] = 0x00030000;          // data_size = 3 (8 bytes/element)
            d1[1] = (int)(16384u << 16); // tensor_dim0 = 16384 elements (low16)
            d1[2] = 0x00010000;          // tensor_dim0 hi=0 | tensor_dim1 = 1
            d1[3] = (int)(16384u << 16); // tile_dim0 = 16384
            d1[4] = 1;                   // tile_dim1 = 1
            d1[5] = 16384;               // tensor_dim0_stride
            d1[6] = 0;
            d1[7] = 0;
            i32x4 dz = {0, 0, 0, 0};
#if defined(__clang_major__) && (__clang_major__ >= 23)
            i32x8 dz8 = {0, 0, 0, 0, 0, 0, 0, 0};        // clang-23 6-arg form
            __builtin_amdgcn_tensor_load_to_lds(d0, d1, dz, dz, dz8, 0);
#else
            __builtin_amdgcn_tensor_load_to_lds(d0, d1, dz, dz, 0);  // clang-22
#endif
            __builtin_amdgcn_s_wait_tensorcnt((short)0);
        }
#else
        {
            const float4* src = (const float4*)(Fpack + g * GRP_FLOATS);
            float4* dst = (float4*)Fsre;
            for (int idx = tid; idx < GRP_FLOATS / 4; idx += 256)
                dst[idx] = src[idx];
        }
#endif
        __syncthreads();

        // ---- stage 1: Vg = X * Fg  (real 256x256 times complex 256x64) ----
        v8f vre[2][4], vim[2][4];
        #pragma unroll
        for (int r = 0; r < 2; ++r)
            #pragma unroll
            for (int c = 0; c < 4; ++c) { vre[r][c] = {}; vim[r][c] = {}; }

        for (int kk = 0; kk < NPIX; kk += 4) {
            const int ks = kk + 2 * hi;
            const int p  = ks >> 1;                 // k-pair index
            const v2f a0 = *(const v2f*)(X + (32 * wv +      lm) * NPIX + ks);
            const v2f a1 = *(const v2f*)(X + (32 * wv + 16 + lm) * NPIX + ks);
            #pragma unroll
            for (int c = 0; c < 4; ++c) {
                const int o = (p * GROUPW + 16 * c + lm) * 2;
                const v2f bre = *(const v2f*)(Fsre + o);   // single ds b64
                const v2f bim = *(const v2f*)(Fsim + o);
                vre[0][c] = wmma4(a0, bre, vre[0][c]);
                vim[0][c] = wmma4(a0, bim, vim[0][c]);
                vre[1][c] = wmma4(a1, bre, vre[1][c]);
                vim[1][c] = wmma4(a1, bim, vim[1][c]);
            }
        }
        // NOTE: no barrier needed here -- the previous group's stage-2 reads
        // of Vs were fenced by the end-of-group barrier.

        // store V tiles pair-planar; {acc[i],acc[i+1]} are adjacent VGPRs
        #pragma unroll
        for (int r = 0; r < 2; ++r)
            #pragma unroll
            for (int c = 0; c < 4; ++c) {
                const int cidx = 16 * c + lm;
                #pragma unroll
                for (int i = 0; i < 8; i += 2) {
                    const int m = 32 * wv + 16 * r + i + 8 * hi;   // even
                    const int o = ((m >> 1) * GROUPW + cidx) * 2;
                    *(float2*)(Vsre + o) = make_float2(vre[r][c][i], vre[r][c][i + 1]);
                    *(float2*)(Vsim + o) = make_float2(vim[r][c][i], vim[r][c][i + 1]);
                }
            }
        __syncthreads();   // Vs visible to all waves before stage 2

        // ---- stage 2: Zg = F * Vg  (complex x complex) ----
        v8f zre[2][4], zim[2][4];
        #pragma unroll
        for (int r = 0; r < 2; ++r)
            #pragma unroll
            for (int c = 0; c < 4; ++c) { zre[r][c] = {}; zim[r][c] = {}; }

        for (int kk = 0; kk < NPIX; kk += 4) {
            const int ks = kk + 2 * hi;
            const int p  = ks >> 1;
            v2f bre[4], bim[4];
            #pragma unroll
            for (int c = 0; c < 4; ++c) {
                const int o = (p * GROUPW + 16 * c + lm) * 2;
                bre[c] = *(const v2f*)(Vsre + o);
                bim[c] = *(const v2f*)(Vsim + o);
            }
            #pragma unroll
            for (int r = 0; r < 2; ++r) {
                const int row = 32 * wv + 16 * r + lm;
                // planar b64 loads: operands arrive pre-packed, pre-negated
                const v2f are = *(const v2f*)(Fre  + row * NPIX + ks);
                const v2f aim = *(const v2f*)(Fim  + row * NPIX + ks);
                const v2f ain = *(const v2f*)(Fimn + row * NPIX + ks);
                #pragma unroll
                for (int c = 0; c < 4; ++c) {
                    zre[r][c] = wmma4(are, bre[c], zre[r][c]);
                    zre[r][c] = wmma4(ain, bim[c], zre[r][c]);
                    zim[r][c] = wmma4(are, bim[c], zim[r][c]);
                    zim[r][c] = wmma4(aim, bre[c], zim[r][c]);
                }
            }
        }

        // ---- |Z|^2 -> radial bins (ds_add_f32 LDS atomics) ----
        #pragma unroll
        for (int r = 0; r < 2; ++r)
            #pragma unroll
            for (int c = 0; c < 4; ++c)
                #pragma unroll
                for (int i = 0; i < 8; ++i) {
                    const float pw = (zre[r][c][i] * zre[r][c][i] +
                                      zim[r][c][i] * zim[r][c][i]) * (float)kPsSc;
                    const int gr = 32 * wv + 16 * r + i + 8 * hi;
                    const int gc = g * GROUPW + 16 * c + lm;
                    atomicAdd(&bins[seg[gr * NPIX + gc]], pw);
                }
        __syncthreads();   // bins complete + Fs/Vs reuse fence for next group
    }

    if (tid < NSEG) binsum[b * NSEG + tid] = bins[tid];
}

// ---------------------------------------------------------------------------
// Kernel 2: counts + weighted log-MSE reduction (tiny, single block)
// ---------------------------------------------------------------------------
__global__ __launch_bounds__(256)
void finalize_kernel(const float* __restrict__ binsum,
                     const int*   __restrict__ seg,
                     const float* __restrict__ mean_cl,
                     const float* __restrict__ weighting,
                     float* __restrict__ out, int Bn) {
    __shared__ int   cnt[NSEG];
    __shared__ float red[256];
    const int tid = threadIdx.x;
    if (tid < NSEG) cnt[tid] = 0;
    __syncthreads();
    for (int i = tid; i < NPIX * NPIX; i += 256) atomicAdd(&cnt[seg[i]], 1);
    __syncthreads();

    float acc = 0.0f;
    const int total = Bn * NBINS;
    for (int idx = tid; idx < total; idx += 256) {
        const int b = idx >> 6;
        const int s = idx & 63;
        const int sg = s + 1;                      // valid segments 1..64
        const float c = (float)cnt[sg];
        const float ps1 = (c > 0.0f) ? binsum[b * NSEG + sg] / c : 0.0f;
        const float d = __logf(ps1 + 1e-12f) - __logf(mean_cl[s] + 1e-12f);
        acc += weighting[s] * d * d;
    }
    red[tid] = acc;
    __syncthreads();
    for (int off = 128; off > 0; off >>= 1) {
        if (tid < off) red[tid] += red[tid + off];
        __syncthreads();
    }
    if (tid == 0) out[0] = red[0] / (float)total;
}

// ---------------------------------------------------------------------------
extern "C" void kernel_launch(void* const* d_in, const int* in_sizes, int n_in,
                              void* d_out, int out_size, void* d_ws, size_t ws_size,
                              hipStream_t stream) {
    const float* pred      = (const float*)d_in[0];
    const float* mean_cl   = (const float*)d_in[1];
    const float* weighting = (const float*)d_in[2];
    const int*   seg       = (const int*)d_in[3];
    const int Bn = in_sizes[0] / (NPIX * NPIX);

    // ws layout: Fre | Fim | Fimn (3 x 256 KB planar) | Fpack (512 KB) | bins
    float* Fre    = (float*)d_ws;
    float* Fim    = Fre  + NPIX * NPIX;
    float* Fimn   = Fim  + NPIX * NPIX;
    float* Fpack  = Fimn + NPIX * NPIX;
    float* binsum = Fpack + NGROUP * GRP_FLOATS;
    (void)ws_size; (void)n_in; (void)out_size;

    build_dft_kernel<<<NPIX, NPIX, 0, stream>>>(Fre, Fim, Fimn, Fpack);

    const size_t shmem = 131072 /*Fs*/ + 131072 /*Vs*/ + 512 /*bins*/;
    fft_ps_kernel<<<Bn, 256, shmem, stream>>>(pred, seg, Fre, Fim, Fimn, Fpack, binsum);

    finalize_kernel<<<1, 256, 0, stream>>>(binsum, seg, mean_cl, weighting,
                                           (float*)d_out, Bn);
}